// RPN_67877663146531
// MI455X (gfx1250) — compile-verified
//
#include <hip/hip_runtime.h>
#include <hip/hip_bf16.h>
#include <stdint.h>

// ---------------------------------------------------------------------------
// RPN proposal generation for MI455X (gfx1250, wave32).
// No matmul exists in this workload -> no WMMA; we target the CDNA5 paths
// that matter here: 320KB LDS sorts, async global->LDS staging (ASYNCcnt),
// and global_prefetch_b8.
// ---------------------------------------------------------------------------

#define NIMG       16
#define NLVL       5
#define ATOT       261888              // sum of anchors over levels
#define KMAX       2000
#define POSTK      1000
#define NEGF       (-1000000000.0f)
#define NMS_TH     0.7f
#define XFORM_CLIP 4.1351665567423560f // log(1000/16)
#define CAND       (NLVL * KMAX)       // 10000 candidate slots per image
#define SORTN      16384               // padded pow2 for final bitonic

__constant__ int c_cnt[NLVL] = {196608, 49152, 12288, 3072, 768};
__constant__ int c_off[NLVL] = {0, 196608, 245760, 258048, 261120};
__constant__ int c_k[NLVL]   = {2000, 2000, 2000, 2000, 768};

// monotonic float <-> uint mapping (descending float == descending uint)
__device__ __forceinline__ unsigned f2u(float f) {
    unsigned u = __float_as_uint(f);
    return (u & 0x80000000u) ? ~u : (u | 0x80000000u);
}
__device__ __forceinline__ float u2f(unsigned u) {
    unsigned b = (u & 0x80000000u) ? (u & 0x7FFFFFFFu) : ~u;
    return __uint_as_float(b);
}

// ---------------------------------------------------------------------------
// Kernel 1: exact top-k threshold via 4-pass MSB radix select.
// One workgroup per (img,lvl). Writes {T, need, 0, 0} into state[g*8..].
// ---------------------------------------------------------------------------
__global__ void __launch_bounds__(256)
rpn_radix_select(const float* __restrict__ obj, unsigned* __restrict__ state) {
    const int g   = blockIdx.x;
    const int img = g / NLVL, lvl = g % NLVL;
    const int cnt = c_cnt[lvl], off = c_off[lvl], K = c_k[lvl];
    const int tid = threadIdx.x;
    const float* src = obj + (size_t)img * ATOT + off;

    __shared__ unsigned hist[256];
    __shared__ unsigned s_prefix, s_need;

    unsigned prefix = 0, need = (unsigned)K;
    for (int pass = 0; pass < 4; ++pass) {
        const int shift = 24 - 8 * pass;
        const unsigned mask_hi = (pass == 0) ? 0u : (0xFFFFFFFFu << (shift + 8));
        hist[tid] = 0;
        __syncthreads();
        for (int i = tid; i < cnt; i += 256) {
            if (i + 4096 < cnt) __builtin_prefetch(src + i + 4096, 0, 1); // global_prefetch_b8
            unsigned u = f2u(src[i]);
            if ((u & mask_hi) == (prefix & mask_hi))
                atomicAdd(&hist[(u >> shift) & 0xFFu], 1u);
        }
        __syncthreads();
        if (tid == 0) {
            unsigned cum = 0;
            int b = 255;
            for (; b >= 0; --b) {
                unsigned c = hist[b];
                if (cum + c >= need) break;
                cum += c;
            }
            if (b < 0) b = 0;
            s_prefix = prefix | ((unsigned)b << shift);
            s_need   = need - cum;
        }
        __syncthreads();
        prefix = s_prefix;
        need   = s_need;
        __syncthreads();
    }
    if (tid == 0) {
        unsigned* st = state + g * 8;
        st[0] = prefix;   // exact k-th key value T
        st[1] = need;     // how many ties (u == T) to take
        st[2] = 0;        // ctr for u > T
        st[3] = 0;        // ctr for u == T
    }
}

// ---------------------------------------------------------------------------
// Kernel 2: compact exactly K winners per (img,lvl).
// ---------------------------------------------------------------------------
__global__ void __launch_bounds__(256)
rpn_compact(const float* __restrict__ obj, unsigned* __restrict__ state,
            int* __restrict__ sel_idx, float* __restrict__ sel_score) {
    const int g   = blockIdx.x;
    const int img = g / NLVL, lvl = g % NLVL;
    const int cnt = c_cnt[lvl], off = c_off[lvl], K = c_k[lvl];
    const int tid = threadIdx.x;
    unsigned* st = state + g * 8;
    const unsigned T = st[0], need = st[1];
    const unsigned cnt_gt = (unsigned)K - need;
    const float* src = obj + (size_t)img * ATOT + off;

    for (int i = tid; i < cnt; i += 256) {
        float f = src[i];
        unsigned u = f2u(f);
        if (u > T) {
            unsigned p = atomicAdd(&st[2], 1u);
            sel_idx[g * KMAX + p]   = i;
            sel_score[g * KMAX + p] = f;
        } else if (u == T) {
            unsigned q = atomicAdd(&st[3], 1u);
            if (q < need) {
                unsigned p = cnt_gt + q;
                sel_idx[g * KMAX + p]   = i;
                sel_score[g * KMAX + p] = f;
            }
        }
    }
}

// ---------------------------------------------------------------------------
// Kernel 3: LDS bitonic sort of the K winners, descending score, ties broken
// by ascending anchor index (matches lax.top_k ordering).
// ---------------------------------------------------------------------------
__global__ void __launch_bounds__(256)
rpn_sort2k(int* __restrict__ sel_idx, float* __restrict__ sel_score) {
    const int g   = blockIdx.x;
    const int lvl = g % NLVL;
    const int K   = c_k[lvl];
    const int tid = threadIdx.x;
    __shared__ unsigned long long kv[2048];

    for (int j = tid; j < 2048; j += 256) {
        if (j < K) {
            unsigned u   = f2u(sel_score[g * KMAX + j]);
            unsigned idx = (unsigned)sel_idx[g * KMAX + j];
            kv[j] = ((unsigned long long)u << 32) | (unsigned long long)(0xFFFFFFFFu - idx);
        } else {
            kv[j] = 0ull; // sinks to the bottom in a descending sort
        }
    }
    __syncthreads();

    for (int kk = 2; kk <= 2048; kk <<= 1) {
        for (int jj = kk >> 1; jj > 0; jj >>= 1) {
            for (int i = tid; i < 2048; i += 256) {
                int l = i ^ jj;
                if (l > i) {
                    unsigned long long a = kv[i], b = kv[l];
                    bool asc = ((i & kk) != 0);        // final merge => descending
                    if ((a > b) == asc) { kv[i] = b; kv[l] = a; }
                }
            }
            __syncthreads();
        }
    }

    for (int j = tid; j < K; j += 256) {
        unsigned long long v = kv[j];
        sel_score[g * KMAX + j] = u2f((unsigned)(v >> 32));
        sel_idx[g * KMAX + j]   = (int)(0xFFFFFFFFu - (unsigned)(v & 0xFFFFFFFFu));
    }
}

// ---------------------------------------------------------------------------
// Kernel 4: decode + clip only the selected boxes (~30x less work than decoding
// every anchor like the reference).
// ---------------------------------------------------------------------------
__global__ void __launch_bounds__(256)
rpn_decode(const float* __restrict__ deltas, const float* __restrict__ anchors,
           const int* __restrict__ sel_idx, float* __restrict__ sel_boxes) {
    const int t = blockIdx.x * 256 + threadIdx.x;
    if (t >= NLVL * NIMG * KMAX) return;
    const int g = t / KMAX, pos = t % KMAX;
    const int img = g / NLVL, lvl = g % NLVL;
    if (pos >= c_k[lvl]) return;

    const int ai = c_off[lvl] + sel_idx[g * KMAX + pos];  // global anchor index
    const float a0 = anchors[ai * 4 + 0], a1 = anchors[ai * 4 + 1];
    const float a2 = anchors[ai * 4 + 2], a3 = anchors[ai * 4 + 3];
    const float w = a2 - a0, h = a3 - a1;
    const float cx = a0 + 0.5f * w, cy = a1 + 0.5f * h;

    const float* d = deltas + ((size_t)img * ATOT + ai) * 4;
    const float dx = d[0], dy = d[1];
    const float dw = fminf(d[2], XFORM_CLIP);
    const float dh = fminf(d[3], XFORM_CLIP);
    const float pcx = dx * w + cx, pcy = dy * h + cy;
    const float pw = __expf(dw) * w, ph = __expf(dh) * h;

    float x0 = pcx - 0.5f * pw, y0 = pcy - 0.5f * ph;
    float x1 = pcx + 0.5f * pw, y1 = pcy + 0.5f * ph;
    x0 = fminf(fmaxf(x0, 0.f), 1024.f); y0 = fminf(fmaxf(y0, 0.f), 1024.f);
    x1 = fminf(fmaxf(x1, 0.f), 1024.f); y1 = fminf(fmaxf(y1, 0.f), 1024.f);

    float* o = sel_boxes + (size_t)(g * KMAX + pos) * 4;
    o[0] = x0; o[1] = y0; o[2] = x1; o[3] = y1;
}

// ---------------------------------------------------------------------------
// Kernel 5: greedy NMS per (img,lvl). Boxes staged into LDS with CDNA5 async
// global->LDS b128 loads (tracked by ASYNCcnt) when the builtin exists.
// The builtin's params are v4i pointers: arg0 = global (AS1), arg1 = LDS (AS3).
// ---------------------------------------------------------------------------
typedef int v4i_t __attribute__((ext_vector_type(4)));
typedef __attribute__((address_space(1))) v4i_t* g4p_t;
typedef __attribute__((address_space(3))) v4i_t* l4p_t;

__global__ void __launch_bounds__(256)
rpn_nms(const float* __restrict__ sel_boxes, float* __restrict__ sel_score) {
    const int g   = blockIdx.x;
    const int lvl = g % NLVL;
    const int K   = c_k[lvl];
    const int tid = threadIdx.x;

    __shared__ float bb[KMAX * 4];      // AoS boxes, 32 KB
    __shared__ float area[KMAX];        // 8 KB
    __shared__ unsigned char keep[KMAX];
    __shared__ unsigned sflag;

    const float* src = sel_boxes + (size_t)g * KMAX * 4;

#if __has_builtin(__builtin_amdgcn_global_load_async_to_lds_b128)
    {
        for (int i = tid; i < K; i += 256) {
            __builtin_amdgcn_global_load_async_to_lds_b128(
                (g4p_t)(src + i * 4), (l4p_t)(bb + i * 4), 0, 0);
        }
  #if __has_builtin(__builtin_amdgcn_s_wait_asynccnt)
        __builtin_amdgcn_s_wait_asynccnt(0);
  #else
        asm volatile("s_wait_asynccnt 0" ::: "memory");
  #endif
    }
#else
    for (int i = tid; i < K * 4; i += 256) bb[i] = src[i];
#endif
    __syncthreads();

    for (int i = tid; i < K; i += 256) {
        area[i] = (bb[i * 4 + 2] - bb[i * 4 + 0]) * (bb[i * 4 + 3] - bb[i * 4 + 1]);
        keep[i] = 0;
    }
    __syncthreads();

    for (int i = 0; i < K; ++i) {
        if (tid == 0) sflag = 0u;
        __syncthreads();
        const float ax0 = bb[i * 4 + 0], ay0 = bb[i * 4 + 1];
        const float ax1 = bb[i * 4 + 2], ay1 = bb[i * 4 + 3];
        const float aa  = area[i];
        unsigned loc = 0;
        for (int j = tid; j < i; j += 256) {
            if (keep[j]) {
                float xx0 = fmaxf(ax0, bb[j * 4 + 0]);
                float yy0 = fmaxf(ay0, bb[j * 4 + 1]);
                float xx1 = fminf(ax1, bb[j * 4 + 2]);
                float yy1 = fminf(ay1, bb[j * 4 + 3]);
                float w = fmaxf(xx1 - xx0, 0.f), h = fmaxf(yy1 - yy0, 0.f);
                float inter = w * h;
                float iou = inter / (aa + area[j] - inter + 1e-9f);
                if (iou > NMS_TH) loc = 1u;
            }
        }
        if (loc) atomicOr(&sflag, 1u);
        __syncthreads();
        if (tid == 0) keep[i] = (sflag == 0u);
        __syncthreads();
    }

    for (int i = tid; i < K; i += 256)
        if (!keep[i]) sel_score[g * KMAX + i] = NEGF;
}

// ---------------------------------------------------------------------------
// Kernel 6: per-image final top-1000 over the 8768 candidates via a 16384-wide
// bitonic sort held entirely in 128 KB of dynamic LDS (fits in the WGP's
// 320 KB). Writes fboxes then fscores into d_out.
// ---------------------------------------------------------------------------
__global__ void __launch_bounds__(256)
rpn_final_topk(const float* __restrict__ sel_score,
               const float* __restrict__ sel_boxes, float* __restrict__ out) {
    extern __shared__ unsigned long long skv[];   // SORTN * 8B = 128 KB
    const int img = blockIdx.x;
    const int tid = threadIdx.x;

    for (int j = tid; j < SORTN; j += 256) {
        unsigned long long v = 0ull;
        if (j < CAND) {
            const int lvl = j / KMAX, pos = j % KMAX;
            if (pos < c_k[lvl]) {
                float f = sel_score[(img * NLVL + lvl) * KMAX + pos];
                v = ((unsigned long long)f2u(f) << 32) |
                    (unsigned long long)(0xFFFFFFFFu - (unsigned)j);
            }
        }
        skv[j] = v;
    }
    __syncthreads();

    for (int kk = 2; kk <= SORTN; kk <<= 1) {
        for (int jj = kk >> 1; jj > 0; jj >>= 1) {
            for (int i = tid; i < SORTN; i += 256) {
                int l = i ^ jj;
                if (l > i) {
                    unsigned long long a = skv[i], b = skv[l];
                    bool asc = ((i & kk) != 0);   // final merge descending
                    if ((a > b) == asc) { skv[i] = b; skv[l] = a; }
                }
            }
            __syncthreads();
        }
    }

    for (int j = tid; j < POSTK; j += 256) {
        unsigned long long v = skv[j];
        unsigned slot = 0xFFFFFFFFu - (unsigned)(v & 0xFFFFFFFFu);
        const int lvl = slot / KMAX, pos = slot % KMAX;
        const float* b = sel_boxes + (size_t)((img * NLVL + lvl) * KMAX + pos) * 4;
        float* ob = out + (size_t)img * POSTK * 4 + (size_t)j * 4;
        ob[0] = b[0]; ob[1] = b[1]; ob[2] = b[2]; ob[3] = b[3];
        out[(size_t)NIMG * POSTK * 4 + img * POSTK + j] = u2f((unsigned)(v >> 32));
    }
}

// ---------------------------------------------------------------------------
extern "C" void kernel_launch(void* const* d_in, const int* in_sizes, int n_in,
                              void* d_out, int out_size, void* d_ws, size_t ws_size,
                              hipStream_t stream) {
    const float* obj     = (const float*)d_in[0];   // [16, 261888]
    const float* deltas  = (const float*)d_in[1];   // [16, 261888, 4]
    const float* anchors = (const float*)d_in[2];   // [261888, 4]
    float* out = (float*)d_out;                     // 64000 boxes + 16000 scores

    char* ws = (char*)d_ws;
    unsigned* state  = (unsigned*)ws;                                 // 80*8 u32
    int*    sel_idx   = (int*)  (ws + 4096);                          // 80*2000
    float*  sel_score = (float*)(ws + 4096 + 80 * KMAX * 4);          // 80*2000
    float*  sel_boxes = (float*)(ws + 4096 + 2 * 80 * KMAX * 4);      // 80*2000*4

    const int G = NIMG * NLVL;  // 80
    rpn_radix_select<<<G, 256, 0, stream>>>(obj, state);
    rpn_compact     <<<G, 256, 0, stream>>>(obj, state, sel_idx, sel_score);
    rpn_sort2k      <<<G, 256, 0, stream>>>(sel_idx, sel_score);
    rpn_decode      <<<(G * KMAX + 255) / 256, 256, 0, stream>>>(deltas, anchors,
                                                                 sel_idx, sel_boxes);
    rpn_nms         <<<G, 256, 0, stream>>>(sel_boxes, sel_score);
    rpn_final_topk  <<<NIMG, 256, SORTN * sizeof(unsigned long long), stream>>>(
        sel_score, sel_boxes, out);
}